// Net_22565758173932
// MI455X (gfx1250) — compile-verified
//
#include <hip/hip_runtime.h>
#include <math.h>

// ---------------------------------------------------------------------------
// 2-layer GCN for gfx1250 (MI455X).  fp32 end-to-end; dense GEMMs use
// V_WMMA_F32_16X16X4_F32; edge aggregation uses L2-resident f32 atomics.
// Working set (~150 MB incl. edge list) fits the 192 MB L2, so the scatter
// phase is L2-bound, not HBM-bound.
// ---------------------------------------------------------------------------

typedef float v2f __attribute__((ext_vector_type(2)));
typedef float v8f __attribute__((ext_vector_type(8)));

#define DIN 128
#define DH  64
#define DOUT 8

// ---------------- utility fill ----------------
__global__ void fill_kernel(float* __restrict__ p, float v, size_t n) {
    size_t i = (size_t)blockIdx.x * blockDim.x + threadIdx.x;
    if (i < n) p[i] = v;
}

// ---------------- degree / rsqrt ----------------
__global__ void degree_kernel(const long long* __restrict__ dst,
                              float* __restrict__ deg, int E) {
    int e = blockIdx.x * blockDim.x + threadIdx.x;
    if (e < E) atomicAdd(&deg[(int)dst[e]], 1.0f);
}

__global__ void rsqrt_kernel(float* __restrict__ deg, int n) {
    int i = blockIdx.x * blockDim.x + threadIdx.x;
    if (i < n) deg[i] = rsqrtf(deg[i]);   // deg >= 1 (self loop)
}

// ---------------- edge prep: int64 -> int32 pair + precomputed norm ---------
__global__ void edge_prep_kernel(const long long* __restrict__ ei,
                                 const float* __restrict__ dis,
                                 int* __restrict__ s32, int* __restrict__ d32,
                                 float* __restrict__ nrm, int E) {
    int e = blockIdx.x * blockDim.x + threadIdx.x;
    if (e >= E) return;
    int s = (int)ei[e];
    int d = (int)ei[(size_t)E + e];
    s32[e] = s;
    d32[e] = d;
    nrm[e] = dis[s] * dis[d];
}

// ---------------- WMMA GEMM 1: h1[N,64] = x[N,128] @ W1[128,64] -------------
// One wave per 16-row block; 4 accumulators cover all 64 output columns so
// the A fragment (x rows) is loaded once and fed to 4 WMMAs per k-step.
// A-frag layout (ISA 7.12.2): lane L holds row (L&15), K = k+(L>>4)*2 .. +1.
__global__ __launch_bounds__(32) void gemm1_wmma(const float* __restrict__ x,
                                                 const float* __restrict__ W1,
                                                 float* __restrict__ h1) {
    const int m0   = blockIdx.x * 16;
    const int lane = threadIdx.x;
    const int r    = lane & 15;
    const int kh   = (lane >> 4) * 2;

    const float* __restrict__ xrow = x + (size_t)(m0 + r) * DIN;
    v8f acc[4] = {};
    for (int k = 0; k < DIN; k += 4) {
        v2f a;
        a.x = xrow[k + kh];
        a.y = xrow[k + kh + 1];
        #pragma unroll
        for (int nt = 0; nt < 4; ++nt) {
            v2f b;
            b.x = W1[(size_t)(k + kh)     * DH + nt * 16 + r];
            b.y = W1[(size_t)(k + kh + 1) * DH + nt * 16 + r];
            acc[nt] = __builtin_amdgcn_wmma_f32_16x16x4_f32(false, a, false, b,
                                                            (short)0, acc[nt],
                                                            false, false);
        }
    }
    const int rowOff = (lane >> 4) * 8;      // lanes 16-31 hold M = v+8
    #pragma unroll
    for (int nt = 0; nt < 4; ++nt)
        #pragma unroll
        for (int v = 0; v < 8; ++v)
            h1[(size_t)(m0 + v + rowOff) * DH + nt * 16 + r] = acc[nt][v];
}

// ---------------- layer-1 scatter (prep path): 16 lanes/edge, float4 -------
__global__ __launch_bounds__(256) void scatter1_prep(const int* __restrict__ s32,
                                                     const int* __restrict__ d32,
                                                     const float* __restrict__ nrm,
                                                     const float* __restrict__ h1,
                                                     float* __restrict__ agg1, int E) {
    int t = blockIdx.x * blockDim.x + threadIdx.x;
    int e = t >> 4;
    if (e >= E) return;
    int j = (t & 15) * 4;
    int s = s32[e], d = d32[e];
    float n = nrm[e];
    float4 v = *(const float4*)(h1 + (size_t)s * DH + j);
    float* dst = agg1 + (size_t)d * DH + j;
    atomicAdd(dst + 0, v.x * n);
    atomicAdd(dst + 1, v.y * n);
    atomicAdd(dst + 2, v.z * n);
    atomicAdd(dst + 3, v.w * n);
}

// ---------------- layer-1 scatter (direct fallback): 32 lanes/edge ----------
__global__ __launch_bounds__(256) void scatter1_direct(const long long* __restrict__ ei,
                                                       const float* __restrict__ dis,
                                                       const float* __restrict__ h1,
                                                       float* __restrict__ agg1, int E) {
    int t = blockIdx.x * blockDim.x + threadIdx.x;
    int e = t >> 5;
    if (e >= E) return;
    int j = (t & 31) * 2;
    int s = (int)ei[e];
    int d = (int)ei[(size_t)E + e];
    float n = dis[s] * dis[d];
    float2 v = *(const float2*)(h1 + (size_t)s * DH + j);
    atomicAdd(&agg1[(size_t)d * DH + j],     v.x * n);
    atomicAdd(&agg1[(size_t)d * DH + j + 1], v.y * n);
}

// ---------------- bias + self-loop + sigmoid (h1 -> h, in place) ------------
__global__ void act1_kernel(const float* __restrict__ agg1,
                            float* __restrict__ h1,       // in: h1, out: h
                            const float* __restrict__ dis,
                            const float* __restrict__ b1, int n /* = N*DH */) {
    int idx = blockIdx.x * blockDim.x + threadIdx.x;
    if (idx >= n) return;
    int i = idx >> 6;
    int f = idx & 63;
    float ds  = dis[i];
    float val = agg1[idx] + h1[idx] * ds * ds + b1[f];
    h1[idx] = 1.0f / (1.0f + __expf(-val));
}

// ---------------- WMMA GEMM 2: h2[N,8] = h[N,64] @ W2[64,8] -----------------
__global__ __launch_bounds__(32) void gemm2_wmma(const float* __restrict__ h,
                                                 const float* __restrict__ W2,
                                                 float* __restrict__ h2) {
    const int m0   = blockIdx.x * 16;
    const int lane = threadIdx.x;
    const int r    = lane & 15;
    const int kh   = (lane >> 4) * 2;

    const float* __restrict__ hrow = h + (size_t)(m0 + r) * DH;
    v8f c = {};
    for (int k = 0; k < DH; k += 4) {
        v2f a, b;
        a.x = hrow[k + kh];
        a.y = hrow[k + kh + 1];
        b.x = (r < DOUT) ? W2[(size_t)(k + kh)     * DOUT + r] : 0.0f;
        b.y = (r < DOUT) ? W2[(size_t)(k + kh + 1) * DOUT + r] : 0.0f;
        c = __builtin_amdgcn_wmma_f32_16x16x4_f32(false, a, false, b,
                                                  (short)0, c, false, false);
    }
    if (r < DOUT) {
        const int rowOff = (lane >> 4) * 8;
        #pragma unroll
        for (int v = 0; v < 8; ++v)
            h2[(size_t)(m0 + v + rowOff) * DOUT + r] = c[v];
    }
}

// ---------------- layer-2 scatter (prep path): 4 lanes/edge, float2 ---------
__global__ __launch_bounds__(256) void scatter2_prep(const int* __restrict__ s32,
                                                     const int* __restrict__ d32,
                                                     const float* __restrict__ nrm,
                                                     const float* __restrict__ h2,
                                                     float* __restrict__ agg2, int E) {
    int t = blockIdx.x * blockDim.x + threadIdx.x;
    int e = t >> 2;
    if (e >= E) return;
    int j = (t & 3) * 2;
    int s = s32[e], d = d32[e];
    float n = nrm[e];
    float2 v = *(const float2*)(h2 + (size_t)s * DOUT + j);
    atomicAdd(&agg2[(size_t)d * DOUT + j],     v.x * n);
    atomicAdd(&agg2[(size_t)d * DOUT + j + 1], v.y * n);
}

// ---------------- layer-2 scatter (direct fallback): 8 lanes/edge -----------
__global__ __launch_bounds__(256) void scatter2_direct(const long long* __restrict__ ei,
                                                       const float* __restrict__ dis,
                                                       const float* __restrict__ h2,
                                                       float* __restrict__ agg2, int E) {
    int t = blockIdx.x * blockDim.x + threadIdx.x;
    int e = t >> 3;
    if (e >= E) return;
    int f = t & 7;
    int s = (int)ei[e];
    int d = (int)ei[(size_t)E + e];
    float n = dis[s] * dis[d];
    atomicAdd(&agg2[(size_t)d * DOUT + f], h2[(size_t)s * DOUT + f] * n);
}

// ---------------- final: z = agg2 + h2 * dis^2 + b2 -------------------------
__global__ void final_kernel(const float* __restrict__ agg2,
                             const float* __restrict__ h2,
                             const float* __restrict__ dis,
                             const float* __restrict__ b2,
                             float* __restrict__ z, int n /* = N*DOUT */) {
    int idx = blockIdx.x * blockDim.x + threadIdx.x;
    if (idx >= n) return;
    int i = idx >> 3;
    int f = idx & 7;
    float ds = dis[i];
    z[idx] = agg2[idx] + h2[idx] * ds * ds + b2[f];
}

// ---------------------------------------------------------------------------
extern "C" void kernel_launch(void* const* d_in, const int* in_sizes, int n_in,
                              void* d_out, int out_size, void* d_ws, size_t ws_size,
                              hipStream_t stream) {
    const float*     x  = (const float*)d_in[0];
    const long long* ei = (const long long*)d_in[1];   // int64 edge_index [2,E]
    const float*     W1 = (const float*)d_in[2];
    const float*     b1 = (const float*)d_in[3];
    const float*     W2 = (const float*)d_in[4];
    const float*     b2 = (const float*)d_in[5];
    float*           z  = (float*)d_out;

    const int N = in_sizes[0] / DIN;       // 100000
    const int E = in_sizes[1] / 2;         // 3200000

    // Workspace layout (floats): deg | agg1 | agg2 | h1 | h2 [| s32 d32 nrm]
    float* deg  = (float*)d_ws;                       // N
    float* agg1 = deg  + (size_t)N;                   // N*64
    float* agg2 = agg1 + (size_t)N * DH;              // N*8
    float* h1   = agg2 + (size_t)N * DOUT;            // N*64
    float* h2   = h1   + (size_t)N * DH;              // N*8
    int*   s32  = (int*)(h2 + (size_t)N * DOUT);      // E
    int*   d32  = s32 + (size_t)E;                    // E
    float* nrm  = (float*)(d32 + (size_t)E);          // E

    const size_t base_bytes = (size_t)N * 145 * sizeof(float);
    const size_t prep_bytes = (size_t)E * 3 * sizeof(float);
    const bool   use_prep   = ws_size >= base_bytes + prep_bytes;

    const int TB = 256;
    // 1) init: deg=1.0 (self loops); agg1+agg2 zeroed (contiguous)
    fill_kernel<<<(N + TB - 1) / TB, TB, 0, stream>>>(deg, 1.0f, (size_t)N);
    size_t nzero = (size_t)N * (DH + DOUT);
    fill_kernel<<<(int)((nzero + TB - 1) / TB), TB, 0, stream>>>(agg1, 0.0f, nzero);

    // 2) degree + rsqrt -> dis (in place in deg)
    degree_kernel<<<(E + TB - 1) / TB, TB, 0, stream>>>(ei + E, deg, E);
    rsqrt_kernel<<<(N + TB - 1) / TB, TB, 0, stream>>>(deg, N);

    // 2b) optional edge prep (int32 endpoints + per-edge norm)
    if (use_prep)
        edge_prep_kernel<<<(E + TB - 1) / TB, TB, 0, stream>>>(ei, deg, s32, d32, nrm, E);

    // 3) h1 = x @ W1 via V_WMMA_F32_16X16X4_F32 (one wave per 16x64 row block)
    gemm1_wmma<<<N / 16, 32, 0, stream>>>(x, W1, h1);

    // 4) layer-1 scatter
    if (use_prep) {
        long long t1 = (long long)E * 16;
        scatter1_prep<<<(int)((t1 + TB - 1) / TB), TB, 0, stream>>>(s32, d32, nrm, h1, agg1, E);
    } else {
        long long t1 = (long long)E * 32;
        scatter1_direct<<<(int)((t1 + TB - 1) / TB), TB, 0, stream>>>(ei, deg, h1, agg1, E);
    }

    // 5) bias + self loop + sigmoid -> h (overwrites h1)
    act1_kernel<<<(N * DH + TB - 1) / TB, TB, 0, stream>>>(agg1, h1, deg, b1, N * DH);

    // 6) h2 = h @ W2 via WMMA (B cols 8..15 zero-padded)
    gemm2_wmma<<<N / 16, 32, 0, stream>>>(h1, W2, h2);

    // 7) layer-2 scatter
    if (use_prep) {
        long long t2 = (long long)E * 4;
        scatter2_prep<<<(int)((t2 + TB - 1) / TB), TB, 0, stream>>>(s32, d32, nrm, h2, agg2, E);
    } else {
        long long t2 = (long long)E * 8;
        scatter2_direct<<<(int)((t2 + TB - 1) / TB), TB, 0, stream>>>(ei, deg, h2, agg2, E);
    }

    // 8) z = agg2 + h2 * dis^2 + b2
    final_kernel<<<(N * DOUT + TB - 1) / TB, TB, 0, stream>>>(agg2, h2, deg, b2, z, N * DOUT);
}